// LoRALinear_28200755265635
// MI455X (gfx1250) — compile-verified
//
#include <hip/hip_runtime.h>

typedef __attribute__((ext_vector_type(2))) float v2f;
typedef __attribute__((ext_vector_type(8))) float v8f;

// Problem shape (matches reference setup_inputs)
constexpr int D_IN  = 4096;
constexpr int D_OUT = 4096;
constexpr int RANK  = 32;
constexpr int M_TOT = 4 * 2048;      // B * S
constexpr float SCALE = 1.0f / 32.0f; // alpha / rank

// ---------------------------------------------------------------------------
// Kernel 1: U[m, r] = SCALE * sum_k x[m,k] * A[k,r]      (M_TOT x RANK, fp32)
// One wave per output row; lane index == rank column. A loads are coalesced
// across the wave, x[m,k] is wave-uniform (L2/L0 broadcast).
// ---------------------------------------------------------------------------
__global__ __launch_bounds__(256) void lora_xa_kernel(
    const float* __restrict__ x,
    const float* __restrict__ A,
    float* __restrict__ U) {
  const int wave = threadIdx.x >> 5;
  const int lane = threadIdx.x & 31;
  const int m = blockIdx.x * 8 + wave;
  const float* __restrict__ xr = x + (size_t)m * D_IN;

  float acc = 0.0f;
#pragma unroll 8
  for (int k = 0; k < D_IN; ++k) {
    acc = fmaf(xr[k], A[k * RANK + lane], acc);
  }
  U[m * RANK + lane] = SCALE * acc;
}

// ---------------------------------------------------------------------------
// Kernel 2: out = x @ W^T  +  U @ Bm  + bias, via V_WMMA_F32_16X16X4_F32.
// Block = 256 threads (8 waves). Block tile 128(M) x 128(N).
// Wave tile 64(M) x 32(N): 4x2 accumulators of 16x16 f32 (8 VGPRs each).
// LoRA handled as a 32-deep K-extension using the same accumulators.
// ---------------------------------------------------------------------------
__global__ __launch_bounds__(256) void lora_gemm_kernel(
    const float* __restrict__ x,     // [M_TOT, D_IN]
    const float* __restrict__ W,     // [D_OUT, D_IN] (row-major => W[n,k] contiguous in k)
    const float* __restrict__ bias,  // [D_OUT]
    const float* __restrict__ Bm,    // [RANK, D_OUT]
    const float* __restrict__ U,     // [M_TOT, RANK]  (already scaled)
    float* __restrict__ out) {       // [M_TOT, D_OUT]
  const int lane  = threadIdx.x & 31;
  const int wave  = threadIdx.x >> 5;
  const int waveM = wave >> 2;              // 0..1
  const int waveN = wave & 3;               // 0..3
  const int m0 = blockIdx.y * 128 + waveM * 64;
  const int n0 = blockIdx.x * 128 + waveN * 32;

  // WMMA f32 16x16x4 per-lane fragment mapping (ISA 7.12.2):
  //   A (16x4): lanes 0-15 row M=lane hold K=0,1; lanes 16-31 hold K=2,3.
  //   B (4x16): lanes 0-15 col N=lane hold K=0,1; lanes 16-31 hold K=2,3.
  const int r  = lane & 15;
  const int kh = (lane >> 4) << 1;          // 0 or 2

  v8f acc[4][2];
#pragma unroll
  for (int i = 0; i < 4; ++i)
#pragma unroll
    for (int j = 0; j < 2; ++j)
      acc[i][j] = (v8f){};

  const float* __restrict__ xp[4];
#pragma unroll
  for (int i = 0; i < 4; ++i)
    xp[i] = x + (size_t)(m0 + i * 16 + r) * D_IN + kh;

  const float* __restrict__ wp[2];
#pragma unroll
  for (int j = 0; j < 2; ++j)
    wp[j] = W + (size_t)(n0 + j * 16 + r) * D_IN + kh;

  // ---- main K loop over D_IN (base GEMM) ----
#pragma unroll 2
  for (int k = 0; k < D_IN; k += 4) {
    v2f a[4], b[2];
#pragma unroll
    for (int i = 0; i < 4; ++i)
      a[i] = *(const v2f*)(xp[i] + k);
#pragma unroll
    for (int j = 0; j < 2; ++j)
      b[j] = *(const v2f*)(wp[j] + k);
#pragma unroll
    for (int i = 0; i < 4; ++i)
#pragma unroll
      for (int j = 0; j < 2; ++j)
        acc[i][j] = __builtin_amdgcn_wmma_f32_16x16x4_f32(
            /*neg_a=*/false, a[i], /*neg_b=*/false, b[j],
            /*c_mod=*/(short)0, acc[i][j],
            /*reuse_a=*/false, /*reuse_b=*/false);
  }

  // ---- LoRA as a K-extension: C += U(16xRANK) * Bm(RANKx16) ----
#pragma unroll
  for (int kr = 0; kr < RANK; kr += 4) {
    v2f a[4], b[2];
#pragma unroll
    for (int i = 0; i < 4; ++i)
      a[i] = *(const v2f*)(U + (size_t)(m0 + i * 16 + r) * RANK + kr + kh);
#pragma unroll
    for (int j = 0; j < 2; ++j) {
      const int n = n0 + j * 16 + r;
      b[j].x = Bm[(size_t)(kr + kh)     * D_OUT + n];
      b[j].y = Bm[(size_t)(kr + kh + 1) * D_OUT + n];
    }
#pragma unroll
    for (int i = 0; i < 4; ++i)
#pragma unroll
      for (int j = 0; j < 2; ++j)
        acc[i][j] = __builtin_amdgcn_wmma_f32_16x16x4_f32(
            false, a[i], false, b[j], (short)0, acc[i][j], false, false);
  }

  // ---- epilogue: + bias, store. C layout: VGPR v, lane l ->
  //   M = v + 8*(l>=16), N = (l & 15). ----
  const int nlo  = lane & 15;
  const int mhi  = (lane >> 4) << 3;        // 0 or 8
  float bv[2];
#pragma unroll
  for (int j = 0; j < 2; ++j)
    bv[j] = bias[n0 + j * 16 + nlo];

#pragma unroll
  for (int i = 0; i < 4; ++i) {
#pragma unroll
    for (int j = 0; j < 2; ++j) {
      const int n = n0 + j * 16 + nlo;
#pragma unroll
      for (int v = 0; v < 8; ++v) {
        const int m = m0 + i * 16 + mhi + v;
        out[(size_t)m * D_OUT + n] = acc[i][j][v] + bv[j];
      }
    }
  }
}

// ---------------------------------------------------------------------------
// Launch
// ---------------------------------------------------------------------------
extern "C" void kernel_launch(void* const* d_in, const int* in_sizes, int n_in,
                              void* d_out, int out_size, void* d_ws, size_t ws_size,
                              hipStream_t stream) {
  const float* x    = (const float*)d_in[0]; // [4,2048,4096]
  const float* W    = (const float*)d_in[1]; // [4096,4096]
  const float* bias = (const float*)d_in[2]; // [4096]
  const float* A    = (const float*)d_in[3]; // [4096,32]
  const float* Bm   = (const float*)d_in[4]; // [32,4096]
  float* out = (float*)d_out;
  float* U   = (float*)d_ws;                 // M_TOT*RANK floats = 1 MB

  (void)in_sizes; (void)n_in; (void)out_size; (void)ws_size;

  // Stage 1: U = SCALE * (x @ A)
  lora_xa_kernel<<<M_TOT / 8, 256, 0, stream>>>(x, A, U);

  // Stage 2: out = x @ W^T + U @ Bm + bias
  dim3 grid(D_OUT / 128, M_TOT / 128);
  lora_gemm_kernel<<<grid, 256, 0, stream>>>(x, W, bias, Bm, U, out);
}